// BellmanFilter_31336081392170
// MI455X (gfx1250) — compile-verified
//
#include <hip/hip_runtime.h>
#include <math.h>

// DFSV Bellman/Kalman information filter, T=8192 sequential steps, state dim 32.
// Single persistent workgroup (256 threads = 8 wave32), all state in LDS.
// 32x32 GEMMs via V_WMMA_F32_16X16X4_F32 (fp32 matches reference precision).
// Woodbury form removes the reference's 128x128 Cholesky entirely.

#define T_STEPS 8192
#define N_OBS   128
#define K_FAC   16
#define SDIM    32
#define LOG2PI  1.8378770664093453f

typedef float v2f __attribute__((ext_vector_type(2)));
typedef float v8f __attribute__((ext_vector_type(8)));

// One wave computes one 16x16 tile of  Cout = Cin + alpha * A * op(B).
// A: (rows x kdim) row-major lda; B: (kdim x cols) ldb, or transposed view.
// f32 WMMA fragment layout: A lanes 0-15 carry M=r with K=(k0,k0+1) in the two
// VGPRs, half-wave 1 carries K=(k0+2,k0+3); B symmetric in N; C/D VGPR j is
// row half*8+j, col = lane&15. Loop trip count is wave-uniform -> EXEC all 1s.
__device__ __forceinline__ void wmma_f32_tile(
    const float* __restrict__ A, int lda,
    const float* __restrict__ B, int ldb, int transB,
    const float* __restrict__ Cin, int ldcin,
    float* __restrict__ Cout, int ldcout,
    int row0, int col0, int kdim, float alpha, int lane)
{
  const int r    = lane & 15;
  const int half = lane >> 4;
  v8f acc = {0.f, 0.f, 0.f, 0.f, 0.f, 0.f, 0.f, 0.f};
  const int kt = kdim >> 2;              // uniform trip count across the wave
  for (int kc = 0; kc < kt; ++kc) {
    const int k0 = kc * 4 + half * 2;
    v2f a, b;
    a[0] = A[(row0 + r) * lda + k0];
    a[1] = A[(row0 + r) * lda + k0 + 1];
    if (transB) {
      b[0] = B[(col0 + r) * ldb + k0];
      b[1] = B[(col0 + r) * ldb + k0 + 1];
    } else {
      b[0] = B[(k0)     * ldb + col0 + r];
      b[1] = B[(k0 + 1) * ldb + col0 + r];
    }
    acc = __builtin_amdgcn_wmma_f32_16x16x4_f32(false, a, false, b,
                                                (short)0, acc, false, false);
  }
#pragma unroll
  for (int j = 0; j < 8; ++j) {
    const int rr = row0 + half * 8 + j;
    const int cc = col0 + r;
    const float base = Cin ? Cin[rr * ldcin + cc] : 0.0f;
    Cout[rr * ldcout + cc] = base + alpha * acc[j];
  }
}

__global__ __launch_bounds__(256)
void dfsv_bellman_filter_kernel(
    const float* __restrict__ obs,     // T x N
    const float* __restrict__ lamg,    // N x K
    const float* __restrict__ Phi_f,   // K x K
    const float* __restrict__ Phi_h,   // K x K
    const float* __restrict__ mug,     // K
    const float* __restrict__ sigma2,  // N
    const float* __restrict__ Qhg,     // K x K
    float* __restrict__ means_out,     // T x 32
    float* __restrict__ infos_out,     // T x 1024
    float* __restrict__ ll_out)        // 1
{
  __shared__ float sLam[N_OBS * K_FAC];      // lambda_r
  __shared__ float sLtR[K_FAC * N_OBS];      // lambda^T / sigma2
  __shared__ float sInvSig[N_OBS];
  __shared__ float sJ[K_FAC * K_FAC];        // Lt_Rinv @ lambda
  __shared__ float sPf[K_FAC * K_FAC];
  __shared__ float sPhm[K_FAC * K_FAC];
  __shared__ float sQh[K_FAC * K_FAC];
  __shared__ float sMu[K_FAC];
  __shared__ float sF[SDIM * SDIM];          // blockdiag(Phi_f, Phi_h)
  __shared__ float Pst[SDIM * SDIM];         // carried posterior covariance
  __shared__ float Ppred[SDIM * SDIM];
  __shared__ float Tmp1[SDIM * SDIM];
  __shared__ float W[SDIM * 2 * SDIM];       // 32x64 Gauss-Jordan workspace
  __shared__ float M16[K_FAC * 2 * K_FAC];   // 16x32 GJ workspace
  __shared__ float Minv[K_FAC * K_FAC];
  __shared__ float N16[K_FAC * K_FAC];       // Minv @ J  (symmetric)
  __shared__ float Hm[SDIM * K_FAC];         // G @ N16
  __shared__ float meanv[SDIM], meanp[SDIM];
  __shared__ float hexp[K_FAC];
  __shared__ float vres[N_OBS];
  __shared__ float uvec[K_FAC], wvec[K_FAC], tvec[K_FAC];
  __shared__ float fac[SDIM];
  __shared__ float red[256];
  __shared__ float PhA[K_FAC * K_FAC];       // stationary P_h
  __shared__ float sLogSig, sLL, sLogDetM, sQuad;

  const int tid  = threadIdx.x;
  const int lane = tid & 31;
  const int wid  = tid >> 5;

  // ---- Phase A: raw constants into LDS --------------------------------
  for (int i = tid; i < N_OBS * K_FAC; i += 256) sLam[i] = lamg[i];
  for (int i = tid; i < K_FAC * K_FAC; i += 256) {
    sPf[i]  = Phi_f[i];
    sPhm[i] = Phi_h[i];
    sQh[i]  = Qhg[i];
  }
  if (tid < K_FAC) sMu[tid] = mug[tid];
  if (tid < N_OBS) sInvSig[tid] = 1.0f / sigma2[tid];
  if (tid == 0) sLL = 0.0f;
  __syncthreads();

  // ---- Phase B: derived constants -------------------------------------
  for (int idx = tid; idx < K_FAC * N_OBS; idx += 256) {
    const int k = idx >> 7, n = idx & 127;
    sLtR[idx] = sLam[n * K_FAC + k] * sInvSig[n];
  }
  for (int idx = tid; idx < SDIM * SDIM; idx += 256) {
    const int i = idx >> 5, j = idx & 31;
    float val = 0.0f;
    if (i < K_FAC && j < K_FAC)            val = sPf[i * K_FAC + j];
    else if (i >= K_FAC && j >= K_FAC)     val = sPhm[(i - K_FAC) * K_FAC + (j - K_FAC)];
    sF[idx] = val;
  }
  if (tid == 0) {
    float s = 0.0f;
    for (int n = 0; n < N_OBS; ++n) s += logf(sigma2[n]);
    sLogSig = s;
  }
  PhA[tid] = sQh[tid];                       // 256 threads == 16x16 exactly
  __syncthreads();

  // J = Lt_Rinv @ lambda  (one element per thread)
  {
    const int k1 = tid >> 4, k2 = tid & 15;
    float acc = 0.0f;
    for (int n = 0; n < N_OBS; ++n)
      acc += sLtR[k1 * N_OBS + n] * sLam[n * K_FAC + k2];
    sJ[tid] = acc;
  }

  // Stationary P_h: fixed point  P <- Phi_h P Phi_h^T + Q_h  (rho^2 ~ 0.90)
  for (int it = 0; it < 320; ++it) {
    const int i = tid >> 4, j = tid & 15;
    float acc = sQh[tid];
    for (int a = 0; a < K_FAC; ++a) {
      const float pa = sPhm[i * K_FAC + a];
      for (int b = 0; b < K_FAC; ++b)
        acc += pa * PhA[a * K_FAC + b] * sPhm[j * K_FAC + b];
    }
    __syncthreads();
    PhA[tid] = acc;
    __syncthreads();
  }

  // Initial carry:  P0 = blockdiag(I, P_h)  (== inv(Omega0)),  mean0 = [0; mu]
  for (int idx = tid; idx < SDIM * SDIM; idx += 256) {
    const int i = idx >> 5, j = idx & 31;
    float val = 0.0f;
    if (i < K_FAC && j < K_FAC)        val = (i == j) ? 1.0f : 0.0f;
    else if (i >= K_FAC && j >= K_FAC) val = PhA[(i - K_FAC) * K_FAC + (j - K_FAC)];
    Pst[idx] = val;
  }
  if (tid < SDIM) meanv[tid] = (tid < K_FAC) ? 0.0f : sMu[tid - K_FAC];
  __syncthreads();

  // =====================  sequential scan over t  =======================
  for (int t = 0; t < T_STEPS; ++t) {
    const float* y = obs + (size_t)t * N_OBS;

    // 1) predicted mean
    if (tid < SDIM) {
      float acc;
      if (tid < K_FAC) {
        acc = 0.0f;
        for (int a = 0; a < K_FAC; ++a) acc += sPf[tid * K_FAC + a] * meanv[a];
      } else {
        const int i = tid - K_FAC;
        acc = sMu[i];
        for (int a = 0; a < K_FAC; ++a)
          acc += sPhm[i * K_FAC + a] * (meanv[K_FAC + a] - sMu[a]);
      }
      meanp[tid] = acc;
    }
    if (tid == 0 && t + 1 < T_STEPS)
      __builtin_prefetch(obs + (size_t)(t + 1) * N_OBS, 0, 0);
    __syncthreads();
    if (tid < K_FAC) hexp[tid] = expf(meanp[K_FAC + tid]);

    // 2) P_pred = F * Pst * F^T + Q            (WMMA f32 16x16x4)
    if (wid < 4)
      wmma_f32_tile(sF, SDIM, Pst, SDIM, 0, nullptr, 0, Tmp1, SDIM,
                    (wid >> 1) * 16, (wid & 1) * 16, SDIM, 1.0f, lane);
    __syncthreads();
    if (wid < 4)
      wmma_f32_tile(Tmp1, SDIM, sF, SDIM, 1, nullptr, 0, Ppred, SDIM,
                    (wid >> 1) * 16, (wid & 1) * 16, SDIM, 1.0f, lane);
    __syncthreads();
    for (int idx = tid; idx < SDIM * SDIM; idx += 256) {
      const int i = idx >> 5, j = idx & 31;
      float q = 0.0f;
      if (i < K_FAC)      { if (i == j) q = hexp[i]; }
      else if (j >= K_FAC)  q = sQh[(i - K_FAC) * K_FAC + (j - K_FAC)];
      Ppred[idx] += q;
    }
    __syncthreads();

    // 3) Omega_pred = inv(P_pred)  via Gauss-Jordan on [P_pred | I]
    for (int idx = tid; idx < SDIM * 2 * SDIM; idx += 256) {
      const int i = idx >> 6, j = idx & 63;
      W[idx] = (j < SDIM) ? Ppred[i * SDIM + j]
                          : (((j - SDIM) == i) ? 1.0f : 0.0f);
    }
    __syncthreads();
    for (int p = 0; p < SDIM; ++p) {
      const float pv = W[p * 64 + p];
      __syncthreads();
      const float pinv = 1.0f / pv;
      if (tid < 64) W[p * 64 + tid] *= pinv;
      if (tid >= 64 && tid < 64 + SDIM) {
        const int i = tid - 64;
        fac[i] = (i == p) ? 0.0f : W[i * 64 + p];
      }
      __syncthreads();
      for (int idx = tid; idx < SDIM * 64; idx += 256) {
        const int i = idx >> 6, j = idx & 63;
        if (i != p) W[idx] -= fac[i] * W[p * 64 + j];
      }
      __syncthreads();
    }

    // infos[t] = Omega_post = Omega_pred + [J 0; 0 0]
    for (int idx = tid; idx < SDIM * SDIM; idx += 256) {
      const int i = idx >> 5, j = idx & 31;
      float val = W[i * 64 + SDIM + j];
      if (i < K_FAC && j < K_FAC) val += sJ[i * K_FAC + j];
      infos_out[(size_t)t * (SDIM * SDIM) + idx] = val;
    }

    // 4) innovation v, quad1 = v^T R^-1 v, u = Lt_Rinv v
    if (tid < N_OBS) {
      float acc = y[tid];
      for (int k = 0; k < K_FAC; ++k) acc -= sLam[tid * K_FAC + k] * meanp[k];
      vres[tid] = acc;
      red[tid]  = acc * acc * sInvSig[tid];
    } else {
      red[tid] = 0.0f;
    }
    __syncthreads();
    for (int s = 128; s > 0; s >>= 1) {
      if (tid < s) red[tid] += red[tid + s];
      __syncthreads();
    }
    if (tid == 0) sQuad = red[0];
    if (tid < K_FAC) {
      float acc = 0.0f;
      for (int n = 0; n < N_OBS; ++n) acc += sLtR[tid * N_OBS + n] * vres[n];
      uvec[tid] = acc;
    }
    __syncthreads();

    // 5) M = I + J * P_ff ; invert (tracks log det M = log det S - sum log s2)
    {
      const int i = tid >> 4, j = tid & 15;
      float acc = (i == j) ? 1.0f : 0.0f;
      for (int a = 0; a < K_FAC; ++a)
        acc += sJ[i * K_FAC + a] * Ppred[a * SDIM + j];
      M16[i * 32 + j]      = acc;
      M16[i * 32 + 16 + j] = (i == j) ? 1.0f : 0.0f;
    }
    if (tid == 0) sLogDetM = 0.0f;
    __syncthreads();
    for (int p = 0; p < K_FAC; ++p) {
      const float pv = M16[p * 32 + p];
      __syncthreads();
      if (tid == 0) sLogDetM += logf(fabsf(pv));
      const float pinv = 1.0f / pv;
      if (tid < 32) M16[p * 32 + tid] *= pinv;
      if (tid >= 32 && tid < 48) {
        const int i = tid - 32;
        fac[i] = (i == p) ? 0.0f : M16[i * 32 + p];
      }
      __syncthreads();
      for (int idx = tid; idx < K_FAC * 32; idx += 256) {
        const int i = idx >> 5, j = idx & 31;
        if (i != p) M16[idx] -= fac[i] * M16[p * 32 + j];
      }
      __syncthreads();
    }
    {
      const int i = tid >> 4, j = tid & 15;
      Minv[tid] = M16[i * 32 + 16 + j];
    }
    __syncthreads();

    // 6) N16 = Minv @ J ; w = Minv u
    {
      const int i = tid >> 4, j = tid & 15;
      float acc = 0.0f;
      for (int a = 0; a < K_FAC; ++a) acc += Minv[i * K_FAC + a] * sJ[a * K_FAC + j];
      N16[tid] = acc;
    }
    if (tid < K_FAC) {
      float acc = 0.0f;
      for (int a = 0; a < K_FAC; ++a) acc += Minv[tid * K_FAC + a] * uvec[a];
      wvec[tid] = acc;
    }
    __syncthreads();
    // t = P_ff w ; H = G @ N16 with G = Ppred[:, :16]
    if (tid < K_FAC) {
      float acc = 0.0f;
      for (int a = 0; a < K_FAC; ++a) acc += Ppred[tid * SDIM + a] * wvec[a];
      tvec[tid] = acc;
    }
    for (int idx = tid; idx < SDIM * K_FAC; idx += 256) {
      const int i = idx >> 4, j = idx & 15;
      float acc = 0.0f;
      for (int a = 0; a < K_FAC; ++a) acc += Ppred[i * SDIM + a] * N16[a * K_FAC + j];
      Hm[idx] = acc;
    }
    __syncthreads();

    // log-likelihood: quad2 = u^T P_ff M^-1 u  (Woodbury)
    if (tid == 0) {
      float q2 = 0.0f;
      for (int a = 0; a < K_FAC; ++a) q2 += uvec[a] * tvec[a];
      sLL += -0.5f * ((float)N_OBS * LOG2PI + sLogSig + sLogDetM + sQuad - q2);
    }

    // 7) P_post = P_pred - H @ G^T   (WMMA, K=16, accumulate base=Ppred)
    if (wid < 4)
      wmma_f32_tile(Hm, K_FAC, Ppred, SDIM, 1, Ppred, SDIM, Pst, SDIM,
                    (wid >> 1) * 16, (wid & 1) * 16, K_FAC, -1.0f, lane);
    __syncthreads();

    // 8) mean_post = mean_pred + P_post[:, :16] @ u
    if (tid < SDIM) {
      float acc = meanp[tid];
      for (int a = 0; a < K_FAC; ++a) acc += Pst[tid * SDIM + a] * uvec[a];
      meanv[tid] = acc;
      means_out[(size_t)t * SDIM + tid] = acc;
    }
    __syncthreads();
  }

  if (tid == 0) ll_out[0] = sLL;
}

extern "C" void kernel_launch(void* const* d_in, const int* in_sizes, int n_in,
                              void* d_out, int out_size, void* d_ws, size_t ws_size,
                              hipStream_t stream) {
  (void)in_sizes; (void)n_in; (void)out_size; (void)d_ws; (void)ws_size;
  const float* obs    = (const float*)d_in[0];
  const float* lamr   = (const float*)d_in[1];
  const float* Phi_f  = (const float*)d_in[2];
  const float* Phi_h  = (const float*)d_in[3];
  const float* mu     = (const float*)d_in[4];
  const float* sigma2 = (const float*)d_in[5];
  const float* Qh     = (const float*)d_in[6];

  float* out       = (float*)d_out;
  float* means_out = out;                                        // T*32
  float* infos_out = out + (size_t)T_STEPS * SDIM;               // T*1024
  float* ll_out    = out + (size_t)T_STEPS * SDIM
                         + (size_t)T_STEPS * SDIM * SDIM;        // 1

  hipLaunchKernelGGL(dfsv_bellman_filter_kernel, dim3(1), dim3(256), 0, stream,
                     obs, lamr, Phi_f, Phi_h, mu, sigma2, Qh,
                     means_out, infos_out, ll_out);
}